// MVDRBeamformer_68607807586621
// MI455X (gfx1250) — compile-verified
//
#include <hip/hip_runtime.h>
#include <math.h>

typedef __attribute__((ext_vector_type(2))) float v2f;
typedef __attribute__((ext_vector_type(8))) float v8f;
typedef int v4i __attribute__((vector_size(16)));

#define B_ 4
#define S_ 2
#define C_ 6
#define F_ 512
#define N_ 2000
#define NC 128
#define LS (NC + 4)   // padded LDS row stride (keeps float4 alignment, spreads banks)

// ---- CDNA5 async global->LDS path (guarded; falls back to sync staging) ----
#if __has_builtin(__builtin_amdgcn_global_load_async_to_lds_b128)
#define HAVE_ASYNC_LDS 1
#else
#define HAVE_ASYNC_LDS 0
#endif

typedef __attribute__((address_space(1))) v4i* gptr_b128;
typedef __attribute__((address_space(3))) v4i* lptr_b128;

__device__ __forceinline__ lptr_b128 to_lds(void* p) {
  // generic LDS pointer: addr[31:0] is the LDS byte offset
  return (lptr_b128)(unsigned int)(unsigned long long)p;
}
__device__ __forceinline__ gptr_b128 to_global(const void* p) {
  return (gptr_b128)(unsigned long long)p;
}

template <int N>
__device__ __forceinline__ void wait_asynccnt() {
#if __has_builtin(__builtin_amdgcn_s_wait_asynccnt)
  __builtin_amdgcn_s_wait_asynccnt(N);
#else
  asm volatile("s_wait_asynccnt %0" :: "i"(N));
#endif
}

// ---------------------------------------------------------------------------
// Kernel A: mask-weighted covariances as an 80x16xK fp32 GEMM per (b,f)
// using V_WMMA_F32_16X16X4_F32, with double-buffered async LDS staging.
// Rows of A: (part in {re,im}) x (s,i) x c = 72 (pad 80).
// Cols of B: [xre_d | xim_d] = 12 (pad 16).
//   R_re[s,i,c,d] = P[re_row][d] + P[im_row][6+d]
//   R_im[s,i,c,d] = P[im_row][d] - P[re_row][6+d]
// ---------------------------------------------------------------------------
__global__ __launch_bounds__(160) void cov_wmma_kernel(
    const float* __restrict__ Xre, const float* __restrict__ Xim,
    const float* __restrict__ masks,
    float* __restrict__ Rre, float* __restrict__ Rim)
{
  __shared__ float sX[2][18 * LS];  // rows 0-5 xre[c], 6-11 xim[c], 12-17 m[s*3+i]
  __shared__ float sP[80 * 16];
  __shared__ float4 sDummy;         // sink for padding async loads (never read)

  const int bf   = blockIdx.x;
  const int b    = bf >> 9;            // F_=512
  const int f    = bf & (F_ - 1);
  const int tid  = threadIdx.x;
  const int wave = tid >> 5;
  const int lane = tid & 31;

  // A-fragment mapping: lanes 0-15 hold rows M with K={k0,k0+1}; lanes 16-31 K={k0+2,k0+3}
  const int rowA = wave * 16 + (lane & 15);
  int arow = -1, mrow = 0;
  if (rowA < 72) {
    int part = rowA / 36;
    int sic  = rowA % 36;
    int s = sic / 18, i = (sic % 18) / 6, c = sic % 6;
    arow = part * 6 + c;          // sX row of x (re rows 0-5, im rows 6-11)
    mrow = 12 + s * 3 + i;        // sX row of the mask
  }
  const int ncol  = lane & 15;
  const int brow  = (ncol < 12) ? ncol : -1;   // sX row index == column index
  const int khalf = (lane >> 4) * 2;

  const int nchunks = (N_ + NC - 1) / NC;   // 16 (last chunk = 80 valid, mult of 4)

  // stage chunk `ch` into buffer `bi`: 18 rows x NC floats (float4 granularity)
  auto stage = [&](int bi, int ch) {
    const int n0 = ch * NC;
    const int nvalid = min(NC, N_ - n0);
    float* buf = &sX[bi][0];
#if HAVE_ASYNC_LDS
    #pragma unroll
    for (int j = 0; j < 4; ++j) {            // exactly 4 async issues per thread
      int idx = tid + j * 160;
      const float* src = Xre;                // safe dummy source
      float* dst = (float*)&sDummy;          // dummy sink
      if (idx < 18 * (NC / 4)) {
        int row = idx >> 5;                  // NC/4 == 32
        int q   = idx & 31;
        const float* base;
        if (row < 6)       base = Xre   + ((size_t)(b * C_ + row)        * F_ + f) * N_;
        else if (row < 12) base = Xim   + ((size_t)(b * C_ + (row - 6))  * F_ + f) * N_;
        else               base = masks + ((size_t)(b * 6  + (row - 12)) * F_ + f) * N_;
        if (q * 4 < nvalid) {
          src = base + n0 + q * 4;
          dst = &buf[row * LS + q * 4];
        } else {
          src = base;                        // in-bounds dummy read
          *(float4*)&buf[row * LS + q * 4] = make_float4(0.f, 0.f, 0.f, 0.f);
        }
      }
      __builtin_amdgcn_global_load_async_to_lds_b128(to_global(src), to_lds(dst), 0, 0);
    }
#else
    for (int idx = tid; idx < 18 * (NC / 4); idx += 160) {
      int row = idx >> 5;
      int q   = idx & 31;
      float4 val = make_float4(0.f, 0.f, 0.f, 0.f);
      if (q * 4 < nvalid) {
        const float* src;
        if (row < 6)       src = Xre   + ((size_t)(b * C_ + row)        * F_ + f) * N_;
        else if (row < 12) src = Xim   + ((size_t)(b * C_ + (row - 6))  * F_ + f) * N_;
        else               src = masks + ((size_t)(b * 6  + (row - 12)) * F_ + f) * N_;
        val = ((const float4*)(src + n0))[q];
      }
      ((float4*)(&buf[row * LS]))[q] = val;
    }
#endif
  };

  v8f acc = {0.f,0.f,0.f,0.f,0.f,0.f,0.f,0.f};

  stage(0, 0);
  for (int ch = 0; ch < nchunks; ++ch) {
    const int cur = ch & 1;
    if (ch + 1 < nchunks) {
      stage(1 - cur, ch + 1);     // prefetch next chunk into the other buffer
#if HAVE_ASYNC_LDS
      wait_asynccnt<4>();         // <=4 outstanding => current chunk's batch done
#endif
    } else {
#if HAVE_ASYNC_LDS
      wait_asynccnt<0>();
#endif
    }
    __syncthreads();

    const float* bufc = &sX[cur][0];
    #pragma unroll 8
    for (int kk = 0; kk < NC; kk += 4) {
      const int k0 = kk + khalf;
      v2f a  = {0.f, 0.f};
      v2f bb = {0.f, 0.f};
      if (arow >= 0) {
        a.x = bufc[mrow * LS + k0]     * bufc[arow * LS + k0];
        a.y = bufc[mrow * LS + k0 + 1] * bufc[arow * LS + k0 + 1];
      }
      if (brow >= 0) {
        bb.x = bufc[brow * LS + k0];
        bb.y = bufc[brow * LS + k0 + 1];
      }
      acc = __builtin_amdgcn_wmma_f32_16x16x4_f32(false, a, false, bb,
                                                  (short)0, acc, false, false);
    }
    __syncthreads();
  }

  // dump D-matrix: lane l, vgpr v -> row = tile*16 + v + 8*(l>>4), col = l&15
  #pragma unroll
  for (int v = 0; v < 8; ++v) {
    int r = wave * 16 + v + 8 * (lane >> 4);
    sP[r * 16 + ncol] = acc[v];
  }
  __syncthreads();

  // combine the 4 real products into complex covariances and store
  for (int t = tid; t < 216; t += 160) {
    int s   = t / 108;
    int rem = t % 108;
    int i   = rem / 36;
    int cd  = rem % 36;
    int c   = cd / 6;
    int d   = cd % 6;
    int sic = s * 18 + i * 6 + c;
    float re = sP[sic * 16 + d]        + sP[(36 + sic) * 16 + 6 + d];
    float im = sP[(36 + sic) * 16 + d] - sP[sic * 16 + 6 + d];
    size_t o = ((((size_t)(b * S_ + s) * 3 + i) * F_ + f) * 36) + (size_t)cd;
    Rre[o] = re;
    Rim[o] = im;
  }
}

// ---------------------------------------------------------------------------
// Kernel B: per-(b,s,f) 6x6 complex GEVD-RTF + MVDR weights (one thread each)
// ---------------------------------------------------------------------------
struct cx { float r, i; };
__device__ __forceinline__ cx cmk(float r, float i){ cx z; z.r=r; z.i=i; return z; }
__device__ __forceinline__ cx cadd(cx a, cx b){ return cmk(a.r+b.r, a.i+b.i); }
__device__ __forceinline__ cx csub(cx a, cx b){ return cmk(a.r-b.r, a.i-b.i); }
__device__ __forceinline__ cx cmul(cx a, cx b){ return cmk(a.r*b.r - a.i*b.i, a.r*b.i + a.i*b.r); }
__device__ __forceinline__ cx cmulc(cx a, cx b){ return cmk(a.r*b.r + a.i*b.i, a.i*b.r - a.r*b.i); } // a*conj(b)
__device__ __forceinline__ cx conjc(cx a){ return cmk(a.r, -a.i); }
__device__ __forceinline__ cx cscale(cx a, float s){ return cmk(a.r*s, a.i*s); }
__device__ __forceinline__ cx cdiv(cx a, cx b){
  float d = fmaxf(b.r*b.r + b.i*b.i, 1e-30f);
  float inv = 1.0f / d;
  return cmk((a.r*b.r + a.i*b.i)*inv, (a.i*b.r - a.r*b.i)*inv);
}

__device__ void chol6(const cx H[6][6], cx L[6][6]) {
  for (int j = 0; j < 6; ++j) {
    float s = H[j][j].r;
    for (int k = 0; k < j; ++k) s -= L[j][k].r*L[j][k].r + L[j][k].i*L[j][k].i;
    float d = sqrtf(fmaxf(s, 1e-20f));
    L[j][j] = cmk(d, 0.f);
    float inv = 1.0f / d;
    for (int i2 = j + 1; i2 < 6; ++i2) {
      cx acc = H[i2][j];
      for (int k = 0; k < j; ++k) acc = csub(acc, cmulc(L[i2][k], L[j][k]));
      L[i2][j] = cscale(acc, inv);
    }
    for (int i2 = 0; i2 < j; ++i2) L[i2][j] = cmk(0.f, 0.f);
  }
}
__device__ void fwd6(const cx L[6][6], const cx in[6], cx out[6]) {   // L y = in
  for (int i2 = 0; i2 < 6; ++i2) {
    cx acc = in[i2];
    for (int k = 0; k < i2; ++k) acc = csub(acc, cmul(L[i2][k], out[k]));
    out[i2] = cscale(acc, 1.0f / L[i2][i2].r);
  }
}
__device__ void bwd6(const cx L[6][6], const cx in[6], cx out[6]) {   // L^H v = in
  for (int i2 = 5; i2 >= 0; --i2) {
    cx acc = in[i2];
    for (int k = i2 + 1; k < 6; ++k) acc = csub(acc, cmul(conjc(L[k][i2]), out[k]));
    out[i2] = cscale(acc, 1.0f / L[i2][i2].r);
  }
}

__global__ __launch_bounds__(128) void mvdr_solve_kernel(
    const float* __restrict__ Rre, const float* __restrict__ Rim,
    float* __restrict__ Wre, float* __restrict__ Wim)
{
  int idx = blockIdx.x * blockDim.x + threadIdx.x;
  if (idx >= B_ * S_ * F_) return;
  int f = idx % F_;
  int s = (idx / F_) % S_;
  int b = idx / (F_ * S_);

  size_t b0 = ((((size_t)(b*S_+s)*3 + 0) * F_) + f) * 36;  // R_tgt
  size_t b1 = ((((size_t)(b*S_+s)*3 + 1) * F_) + f) * 36;  // R_noise_1
  size_t b2 = ((((size_t)(b*S_+s)*3 + 2) * F_) + f) * 36;  // R_noise_2

  cx A[6][6], H[6][6], L[6][6], Z[6][6];
  for (int c = 0; c < 6; ++c)
    for (int d = 0; d < 6; ++d) {
      A[c][d] = cmk(Rre[b0 + c*6 + d], Rim[b0 + c*6 + d]);
      H[c][d] = cmk(Rre[b2 + c*6 + d], Rim[b2 + c*6 + d]);
    }
  for (int c = 0; c < 6; ++c) H[c][c].r += 1e-5f;
  chol6(H, L);

  // T = L^{-1} A   (in place)
  for (int col = 0; col < 6; ++col)
    for (int i2 = 0; i2 < 6; ++i2) {
      cx acc = A[i2][col];
      for (int k = 0; k < i2; ++k) acc = csub(acc, cmul(L[i2][k], A[k][col]));
      A[i2][col] = cscale(acc, 1.0f / L[i2][i2].r);
    }
  // Z = L^{-1} T^H ;  whitened matrix C = Z^H (PSD)
  for (int col = 0; col < 6; ++col) {
    cx rhs[6], y[6];
    for (int i2 = 0; i2 < 6; ++i2) rhs[i2] = conjc(A[col][i2]);
    fwd6(L, rhs, y);
    for (int i2 = 0; i2 < 6; ++i2) Z[i2][col] = y[i2];
  }
  // power iteration on C (top eigenvector of PSD matrix == eigh's last column)
  cx u[6];
  for (int i2 = 0; i2 < 6; ++i2) u[i2] = cmk(1.f, 0.f);
  for (int it = 0; it < 60; ++it) {
    cx w[6];
    float nrm = 0.f;
    for (int i2 = 0; i2 < 6; ++i2) {
      cx acc = cmk(0.f, 0.f);
      for (int j2 = 0; j2 < 6; ++j2) acc = cadd(acc, cmul(conjc(Z[j2][i2]), u[j2]));
      w[i2] = acc;
      nrm += acc.r*acc.r + acc.i*acc.i;
    }
    float inv = rsqrtf(fmaxf(nrm, 1e-30f));
    for (int i2 = 0; i2 < 6; ++i2) u[i2] = cscale(w[i2], inv);
  }
  // v = L^{-H} u ;  rtf = v / v[0]
  cx v[6], rtf[6];
  bwd6(L, u, v);
  for (int i2 = 0; i2 < 6; ++i2) rtf[i2] = cdiv(v[i2], v[0]);

  // a = (R_noise_1 + load I)^{-1} rtf ;  w = a / max(Re(rtf^H a), eps)
  for (int c = 0; c < 6; ++c)
    for (int d = 0; d < 6; ++d)
      H[c][d] = cmk(Rre[b1 + c*6 + d], Rim[b1 + c*6 + d]);
  for (int c = 0; c < 6; ++c) H[c][c].r += 1e-5f;
  chol6(H, L);
  cx y2[6], a2[6];
  fwd6(L, rtf, y2);
  bwd6(L, y2, a2);
  float denom = 0.f;
  for (int i2 = 0; i2 < 6; ++i2) denom += rtf[i2].r*a2[i2].r + rtf[i2].i*a2[i2].i;
  denom = fmaxf(denom, 1e-6f);
  float invd = 1.0f / denom;
  size_t wb = ((size_t)(b*S_ + s) * F_ + f) * 6;
  for (int c = 0; c < 6; ++c) {
    Wre[wb + c] = a2[c].r * invd;
    Wim[wb + c] = a2[c].i * invd;
  }
}

// ---------------------------------------------------------------------------
// Kernel C: apply beamformer Y[b,s,f,n] = sum_c X[c,n] * conj(w[s,c]),
// output interleaved (re, im) on last axis. Memory-bound, float4 streaming.
// ---------------------------------------------------------------------------
__global__ __launch_bounds__(256) void apply_kernel(
    const float* __restrict__ Xre, const float* __restrict__ Xim,
    const float* __restrict__ Wre, const float* __restrict__ Wim,
    float* __restrict__ out)
{
  const int bf = blockIdx.x;
  const int b  = bf >> 9;
  const int f  = bf & (F_ - 1);

  float wr[2][6], wi[2][6];
  for (int s = 0; s < 2; ++s) {
    size_t wb = ((size_t)(b*S_ + s) * F_ + f) * 6;
    for (int c = 0; c < 6; ++c) { wr[s][c] = Wre[wb + c]; wi[s][c] = Wim[wb + c]; }
  }
  const float* xr[6];
  const float* xi[6];
  for (int c = 0; c < 6; ++c) {
    size_t xb = ((size_t)(b*C_ + c) * F_ + f) * N_;
    xr[c] = Xre + xb;
    xi[c] = Xim + xb;
  }
  for (int g = threadIdx.x; g < N_ / 4; g += blockDim.x) {
    float4 yre[2], yim[2];
    for (int s = 0; s < 2; ++s) {
      yre[s] = make_float4(0.f,0.f,0.f,0.f);
      yim[s] = make_float4(0.f,0.f,0.f,0.f);
    }
    #pragma unroll
    for (int c = 0; c < 6; ++c) {
      float4 r4 = ((const float4*)xr[c])[g];
      float4 i4 = ((const float4*)xi[c])[g];
      #pragma unroll
      for (int s = 0; s < 2; ++s) {
        float a = wr[s][c], bw = wi[s][c];
        yre[s].x += r4.x*a + i4.x*bw;  yim[s].x += i4.x*a - r4.x*bw;
        yre[s].y += r4.y*a + i4.y*bw;  yim[s].y += i4.y*a - r4.y*bw;
        yre[s].z += r4.z*a + i4.z*bw;  yim[s].z += i4.z*a - r4.z*bw;
        yre[s].w += r4.w*a + i4.w*bw;  yim[s].w += i4.w*a - r4.w*bw;
      }
    }
    #pragma unroll
    for (int s = 0; s < 2; ++s) {
      size_t ob = (((size_t)(b*S_ + s) * F_ + f) * N_) * 2 + (size_t)g * 8;
      ((float4*)(out + ob))[0] = make_float4(yre[s].x, yim[s].x, yre[s].y, yim[s].y);
      ((float4*)(out + ob))[1] = make_float4(yre[s].z, yim[s].z, yre[s].w, yim[s].w);
    }
  }
}

// ---------------------------------------------------------------------------
extern "C" void kernel_launch(void* const* d_in, const int* in_sizes, int n_in,
                              void* d_out, int out_size, void* d_ws, size_t ws_size,
                              hipStream_t stream) {
  (void)in_sizes; (void)n_in; (void)out_size; (void)ws_size;
  const float* Xre   = (const float*)d_in[0];
  const float* Xim   = (const float*)d_in[1];
  const float* masks = (const float*)d_in[2];
  float* out = (float*)d_out;

  float* Rre = (float*)d_ws;                         // B*S*3*F*36 = 442368
  float* Rim = Rre + (size_t)B_*S_*3*F_*36;
  float* Wre = Rim + (size_t)B_*S_*3*F_*36;          // B*S*F*6 = 24576
  float* Wim = Wre + (size_t)B_*S_*F_*6;

  cov_wmma_kernel<<<B_*F_, 160, 0, stream>>>(Xre, Xim, masks, Rre, Rim);
  mvdr_solve_kernel<<<(B_*S_*F_ + 127)/128, 128, 0, stream>>>(Rre, Rim, Wre, Wim);
  apply_kernel<<<B_*F_, 256, 0, stream>>>(Xre, Xim, Wre, Wim, out);
}